// MatrixTree_27865747816847
// MI455X (gfx1250) — compile-verified
//
#include <hip/hip_runtime.h>
#include <hip/hip_bf16.h>

typedef __attribute__((ext_vector_type(2))) float v2f;
typedef __attribute__((ext_vector_type(8))) float v8f;

#define NN     256
#define LDA    257   // padded row stride of main matrix (banks rotate)
#define LDF    17    // padded stride of (negated) column-panel snapshot
#define LDD    17    // Dinv stride
#define LDAUG  33    // augmented 16x32 block stride
#define NB     16    // pivot block size
#define NBLK   16    // number of pivot blocks
#define EPSF   1e-5f

// smem floats: 256*257 + 256*17 + 16*33 + 16*17 + 256 + 256 = 71456 -> 285824 B
#define SMEM_FLOATS (NN*LDA + NN*LDF + NB*LDAUG + NB*LDD + NN + NN)

__global__ __launch_bounds__(256)
void MatrixTree_27865747816847_kernel(const float* __restrict__ x,
                                      float* __restrict__ out) {
    extern __shared__ float smem[];
    float* Am      = smem;                    // [NN][LDA]  Laplacian -> inverse, in place
    float* Fbuf    = Am     + NN * LDA;       // [NN][LDF]  NEGATED column-panel snapshot
    float* Daug    = Fbuf   + NN * LDF;       // [NB][LDAUG] augmented pivot block
    float* Dinv    = Daug   + NB * LDAUG;     // [NB][LDD]
    float* invdiag = Dinv   + NB * LDD;       // [NN]
    float* rootw   = invdiag + NN;            // [NN] diag_exp (root scores)

    const int tid  = threadIdx.x;
    const int b    = blockIdx.x;
    const float* xg = x   + (size_t)b * NN * NN;
    float*       og = out + (size_t)b * NN * NN;

    // ---------------- Phase 1: build Laplacian in LDS (vectorized HBM pass) ----------------
    {
        const float4* xg4 = (const float4*)xg;
        for (int q = tid; q < (NN * NN) / 4; q += 256) {
            float4 v4 = xg4[q];                       // global_load_b128, coalesced
            int idx = q * 4;
            int i = idx >> 8, j = idx & 255;
            float* row = Am + i * LDA + j;
            row[0] = __expf(v4.x) + EPSF;
            row[1] = __expf(v4.y) + EPSF;
            row[2] = __expf(v4.z) + EPSF;
            row[3] = __expf(v4.w) + EPSF;
        }
    }
    __syncthreads();
    {   // one thread per column j: colsum, negate off-diag, set diagonal
        int j = tid;
        float dgv = Am[j * LDA + j];          // exp(x[j][j]) + EPS
        rootw[j] = dgv - EPSF;                // diag_exp[j]
        float s = 0.f;
        for (int i = 0; i < NN; ++i) {
            if (i != j) {
                float v = Am[i * LDA + j];
                s += v;
                Am[i * LDA + j] = -v;
            }
        }
        Am[j * LDA + j] = s;
    }
    __syncthreads();
    Am[tid] = rootw[tid];                     // row 0 <- diag_exp (root row)
    __syncthreads();

    const int lane = tid & 31;
    const int wave = tid >> 5;
    const int half = (lane >> 4) & 1;
    const int l15  = lane & 15;

    // ---------------- Phase 2: blocked in-place Gauss-Jordan inverse ----------------
    for (int kb = 0; kb < NBLK; ++kb) {
        const int k0 = kb * NB;

        // (a) snapshot NEGATED column panel: Fbuf = -A[:, k0:k0+16]  (one row per thread)
        #pragma unroll
        for (int c = 0; c < NB; ++c)
            Fbuf[tid * LDF + c] = -Am[tid * LDA + k0 + c];

        // (b) invert 16x16 pivot block via augmented GJ (threads 0..15 own rows)
        if (tid < NB) {
            #pragma unroll
            for (int c = 0; c < NB; ++c) Daug[tid * LDAUG + c] = Am[(k0 + tid) * LDA + k0 + c];
            #pragma unroll
            for (int c = 0; c < NB; ++c) Daug[tid * LDAUG + NB + c] = (tid == c) ? 1.f : 0.f;
        }
        __syncthreads();
        for (int k = 0; k < NB; ++k) {
            if (tid == k) {
                float p = 1.f / Daug[k * LDAUG + k];
                #pragma unroll
                for (int c = 0; c < 2 * NB; ++c) Daug[k * LDAUG + c] *= p;
            }
            __syncthreads();
            if (tid < NB && tid != k) {
                float f = Daug[tid * LDAUG + k];
                #pragma unroll
                for (int c = 0; c < 2 * NB; ++c) Daug[tid * LDAUG + c] -= f * Daug[k * LDAUG + c];
            }
            __syncthreads();
        }
        if (tid < NB) {
            #pragma unroll
            for (int c = 0; c < NB; ++c) Dinv[tid * LDD + c] = Daug[tid * LDAUG + NB + c];
        }
        __syncthreads();

        // (c) row panel: A[k0:k0+16, j] <- Dinv * A_old[k0:k0+16, j], j outside block
        {
            int j = tid;
            if (j < k0 || j >= k0 + NB) {
                float v[NB];
                #pragma unroll
                for (int c = 0; c < NB; ++c) v[c] = Am[(k0 + c) * LDA + j];
                #pragma unroll
                for (int r = 0; r < NB; ++r) {
                    float s = 0.f;
                    #pragma unroll
                    for (int c = 0; c < NB; ++c) s += Dinv[r * LDD + c] * v[c];
                    Am[(k0 + r) * LDA + j] = s;
                }
            }
        }
        __syncthreads();

        // (e/f) column panel: A[i, k0:] <- -F_i*Dinv == (negF_i)*Dinv ; pivot block <- Dinv
        {
            int i = tid;
            int r = i - k0;
            if (r >= 0 && r < NB) {
                #pragma unroll
                for (int c = 0; c < NB; ++c) Am[i * LDA + k0 + c] = Dinv[r * LDD + c];
            } else {
                float f[NB];                         // f = -F (already negated in Fbuf)
                #pragma unroll
                for (int c = 0; c < NB; ++c) f[c] = Fbuf[i * LDF + c];
                #pragma unroll
                for (int c = 0; c < NB; ++c) {
                    float s = 0.f;
                    #pragma unroll
                    for (int cc = 0; cc < NB; ++cc) s += f[cc] * Dinv[cc * LDD + c];
                    Am[i * LDA + k0 + c] = s;        // = -F_i * Dinv
                }
            }
        }

        // (d) trailing update C <- C + (-F)*R via fp32 WMMA 16x16x4 (K=16 -> 4 chained).
        // Work items: 15 block-cols x 2 row-halves = 30, strided over 8 waves.
        // B operand (row-panel block) loaded ONCE per item and reused over 7-8 row tiles.
        for (int it = wave; it < 30; it += 8) {
            const int bjc = it >> 1;
            const int hh  = it & 1;
            const int bj  = bjc + (bjc >= kb);
            const int col0 = bj * NB;

            v2f Bv[4];                                // B 4x16 chunks: k = kc+v+2*half, n = l15
            #pragma unroll
            for (int kc4 = 0; kc4 < 4; ++kc4) {
                #pragma unroll
                for (int v = 0; v < 2; ++v)
                    Bv[kc4][v] = Am[(k0 + kc4 * 4 + v + 2 * half) * LDA + col0 + l15];
            }

            const int biBeg = hh ? 8 : 0;
            const int biEnd = hh ? 15 : 8;
            for (int bic = biBeg; bic < biEnd; ++bic) {
                const int bi = bic + (bic >= kb);
                const int row0 = bi * NB;

                v8f acc;
                #pragma unroll
                for (int v = 0; v < 8; ++v)           // C/D layout: m = v + 8*half, n = l15
                    acc[v] = Am[(row0 + v + 8 * half) * LDA + col0 + l15];

                v2f Av[4];                            // A 16x4 chunks: m = l15, k = kc+v+2*half
                #pragma unroll
                for (int kc4 = 0; kc4 < 4; ++kc4) {
                    #pragma unroll
                    for (int v = 0; v < 2; ++v)
                        Av[kc4][v] = Fbuf[(row0 + l15) * LDF + kc4 * 4 + v + 2 * half];
                }

                #pragma unroll
                for (int kc4 = 0; kc4 < 4; ++kc4)
                    acc = __builtin_amdgcn_wmma_f32_16x16x4_f32(
                            /*neg_a=*/false, Av[kc4], /*neg_b=*/false, Bv[kc4],
                            /*c_mod=*/(short)0, acc, /*reuse_a=*/false, /*reuse_b=*/false);

                #pragma unroll
                for (int v = 0; v < 8; ++v)
                    Am[(row0 + v + 8 * half) * LDA + col0 + l15] = acc[v];
            }
        }
        __syncthreads();
    }

    // ---------------- Phase 3: epilogue (vectorized HBM pass) ----------------
    invdiag[tid] = Am[tid * LDA + tid];
    __syncthreads();
    {
        const float4* xg4 = (const float4*)xg;
        float4*       og4 = (float4*)og;
        for (int q = tid; q < (NN * NN) / 4; q += 256) {
            float4 v4 = xg4[q];                      // re-read x, coalesced b128
            int idx = q * 4;
            int i = idx >> 8, jb = idx & 255;
            float ex[4] = { __expf(v4.x), __expf(v4.y), __expf(v4.z), __expf(v4.w) };
            float4 o4;
            float* op = &o4.x;
            #pragma unroll
            for (int u = 0; u < 4; ++u) {
                int j = jb + u;
                float t1 = (j == 0) ? 0.f : ex[u] * invdiag[j];        // exp_x * inv_diag[j]
                float t2 = (i == 0) ? 0.f : ex[u] * Am[j * LDA + i];   // exp_x * inv^T
                float o = t1 - t2;
                if (i == j) o += rootw[i] * Am[i * LDA + 0];           // + diag(diag_exp*inv[:,0])
                op[u] = o;
            }
            og4[q] = o4;                             // global_store_b128, coalesced
        }
    }
}

extern "C" void kernel_launch(void* const* d_in, const int* in_sizes, int n_in,
                              void* d_out, int out_size, void* d_ws, size_t ws_size,
                              hipStream_t stream) {
    const float* x = (const float*)d_in[0];
    float* out = (float*)d_out;
    (void)in_sizes; (void)n_in; (void)out_size; (void)d_ws; (void)ws_size;

    const size_t smem_bytes = (size_t)SMEM_FLOATS * sizeof(float); // 285,824 B < 320 KB/WGP
    hipFuncSetAttribute((const void*)MatrixTree_27865747816847_kernel,
                        hipFuncAttributeMaxDynamicSharedMemorySize, (int)smem_bytes);
    MatrixTree_27865747816847_kernel<<<dim3(NN), dim3(256), smem_bytes, stream>>>(x, out);
}